// GMMAttention_22737556865192
// MI455X (gfx1250) — compile-verified
//
#include <hip/hip_runtime.h>
#include <hip/hip_bf16.h>
#include <stdint.h>

// Problem constants (from reference)
#define BSZ   32
#define TLEN  4000
#define ENC   512
#define RNN   1024
#define ATT   128
#define MIX   5
#define EPSV  1e-10f
#define TWO_PI 6.283185307179586f

typedef __attribute__((ext_vector_type(2))) float v2f;
typedef __attribute__((ext_vector_type(8))) float v8f;

__device__ __forceinline__ float softplus_f(float x) {
    // numerically stable: max(x,0) + log1p(exp(-|x|))
    return fmaxf(x, 0.0f) + log1pf(expf(-fabsf(x)));
}

// ---------------------------------------------------------------------------
// Kernel A: one workgroup (16 waves). WMMA fp32 GEMM pq = h @ Wq^T + bq,
// tanh, tiny second GEMM, then per-batch GMM parameter computation.
// params[b*16 + m]      = w/(z+eps)         (coef)
// params[b*16 + 5 + m]  = mu + softplus(dh) (mu_new)
// params[b*16 + 10 + m] = 1/(2*sigma^2+eps) (inv denom)
// ---------------------------------------------------------------------------
__global__ __launch_bounds__(512) void gmm_params_kernel(
    const float* __restrict__ h,          // (32,1024)
    const float* __restrict__ Wq,         // (128,1024)
    const float* __restrict__ bq,         // (128)
    const float* __restrict__ Wv,         // (15,128)
    const float* __restrict__ mu,         // (32,1,5)
    const float* __restrict__ delta_bias, // (1,5)
    const float* __restrict__ sigma_bias, // (1,5)
    float* __restrict__ params)           // (32,16)
{
    __shared__ float tq[BSZ][ATT];     // tanh(pq), 16 KB
    __shared__ float inter_s[BSZ][16]; // 15 used per row

    const int tid  = threadIdx.x;
    const int wave = tid >> 5;   // 0..15
    const int lane = tid & 31;
    const int mt   = wave >> 3;  // row tile of pq (batch dim), 0..1
    const int nt   = wave & 7;   // col tile (att dim), 0..7
    const int lr   = lane & 15;
    const int lhi  = lane >> 4;  // 0 or 1

    // A-frag: A[m][k0], A[m][k0+1]  with m = lr, k0 = 2*lhi  (16x4 f32 layout)
    // B-frag: B[k0][n], B[k0+1][n]  with n = lr  -> B[k][n] = Wq[n][k]
    const float2* a2 = (const float2*)(h  + (size_t)(mt * 16 + lr) * RNN);
    const float2* b2 = (const float2*)(Wq + (size_t)(nt * 16 + lr) * RNN);

    v8f c = {};
    #pragma unroll 8
    for (int k2 = 0; k2 < RNN / 2; k2 += 2) {    // k advances by 4 per WMMA
        float2 af = a2[k2 + lhi];
        float2 bf = b2[k2 + lhi];
        v2f a; a.x = af.x; a.y = af.y;
        v2f b; b.x = bf.x; b.y = bf.y;
        c = __builtin_amdgcn_wmma_f32_16x16x4_f32(
                /*neg_a=*/false, a, /*neg_b=*/false, b,
                /*c_mod=*/(short)0, c, /*reuse_a=*/false, /*reuse_b=*/false);
    }

    // C layout: VGPR r -> M = r + 8*lhi, N = lr
    {
        const int n = nt * 16 + lr;
        const float bias = bq[n];
        #pragma unroll
        for (int r = 0; r < 8; ++r) {
            const int m = mt * 16 + r + 8 * lhi;
            tq[m][n] = tanhf(c[r] + bias);
        }
    }
    __syncthreads();

    // inter[b][j] = sum_k tq[b][k] * Wv[j][k]   (32 x 15)
    if (tid < BSZ * 15) {
        const int b = tid / 15, j = tid % 15;
        float s = 0.0f;
        #pragma unroll 8
        for (int k = 0; k < ATT; ++k) s += tq[b][k] * Wv[j * ATT + k];
        inter_s[b][j] = s;
    }
    __syncthreads();

    // per-batch GMM params
    if (tid < BSZ) {
        const int b = tid;
        float wh[MIX];
        float mx = -3.0e38f;
        #pragma unroll
        for (int m = 0; m < MIX; ++m) { wh[m] = inter_s[b][m]; mx = fmaxf(mx, wh[m]); }
        float sum = 0.0f;
        #pragma unroll
        for (int m = 0; m < MIX; ++m) { wh[m] = expf(wh[m] - mx); sum += wh[m]; }
        const float inv_sum = 1.0f / sum;
        #pragma unroll
        for (int m = 0; m < MIX; ++m) {
            const float w      = wh[m] * inv_sum;
            const float dh     = inter_s[b][5 + m]  + delta_bias[m];
            const float sh     = inter_s[b][10 + m] + sigma_bias[m];
            const float delta  = softplus_f(dh);
            const float sig    = softplus_f(sh);
            const float sig2   = sig * sig;
            const float z      = sqrtf(TWO_PI * sig2);
            const float mu_new = mu[b * MIX + m] + delta;
            params[b * 16 + m]      = w / (z + EPSV);
            params[b * 16 + 5 + m]  = mu_new;
            params[b * 16 + 10 + m] = 1.0f / (2.0f * sig2 + EPSV);
        }
    }
}

// ---------------------------------------------------------------------------
// Kernel B: att[b][t] = mask ? sum_m coef*exp(-(t-mu)^2*inv) : 0
// ---------------------------------------------------------------------------
__global__ __launch_bounds__(256) void att_kernel(
    const uint8_t* __restrict__ mask,   // (32,4000) bool
    const float*   __restrict__ params, // (32,16)
    float*         __restrict__ att)    // (32,4000)
{
    const int b = blockIdx.y;
    const int t = blockIdx.x * 256 + threadIdx.x;
    if (t >= TLEN) return;
    const float tf = (float)t;
    float e = 0.0f;
    #pragma unroll
    for (int m = 0; m < MIX; ++m) {
        const float coef = params[b * 16 + m];
        const float muv  = params[b * 16 + 5 + m];
        const float inv  = params[b * 16 + 10 + m];
        const float d = tf - muv;
        e += coef * __expf(-d * d * inv);
    }
    att[b * TLEN + t] = mask[b * TLEN + t] ? e : 0.0f;
}

// ---------------------------------------------------------------------------
// Kernel C: streaming weighted reduction (the memory-bound hot loop).
// Grid (TCHUNKS, B), 128 threads; each thread owns one float4 column slice,
// loops over 125 timesteps -> perfectly coalesced 2 KB/row global_load_b128.
// ---------------------------------------------------------------------------
#define TCHUNKS 32
#define CHLEN   (TLEN / TCHUNKS)   // 125

__global__ __launch_bounds__(128) void ctx_partial_kernel(
    const float* __restrict__ inputs,   // (32,4000,512)
    const float* __restrict__ att,      // (32,4000)
    float*       __restrict__ partials) // (32,32,512)
{
    const int b  = blockIdx.y;
    const int ch = blockIdx.x;
    const int d4 = threadIdx.x;                 // 0..127 (float4 lanes over 512)
    const int t0 = ch * CHLEN;

    const float4* in4 = (const float4*)(inputs + (size_t)b * TLEN * ENC)
                        + (size_t)t0 * (ENC / 4) + d4;
    const float* ab = att + b * TLEN + t0;

    float4 acc; acc.x = 0.f; acc.y = 0.f; acc.z = 0.f; acc.w = 0.f;
    #pragma unroll 5
    for (int t = 0; t < CHLEN; ++t) {
        const float w = ab[t];
        const float4 v = in4[(size_t)t * (ENC / 4)];
        acc.x = fmaf(w, v.x, acc.x);
        acc.y = fmaf(w, v.y, acc.y);
        acc.z = fmaf(w, v.z, acc.z);
        acc.w = fmaf(w, v.w, acc.w);
    }
    float4* p4 = (float4*)partials;
    p4[((size_t)b * TCHUNKS + ch) * (ENC / 4) + d4] = acc;
}

// ---------------------------------------------------------------------------
// Kernel D: deterministic chunk reduction -> context (32,512)
// ---------------------------------------------------------------------------
__global__ __launch_bounds__(512) void ctx_reduce_kernel(
    const float* __restrict__ partials, // (32,32,512)
    float*       __restrict__ out)      // (32,512)
{
    const int b = blockIdx.x;
    const int d = threadIdx.x;
    float s = 0.0f;
    #pragma unroll 8
    for (int c = 0; c < TCHUNKS; ++c)
        s += partials[((size_t)b * TCHUNKS + c) * ENC + d];
    out[b * ENC + d] = s;
}

// ---------------------------------------------------------------------------
extern "C" void kernel_launch(void* const* d_in, const int* in_sizes, int n_in,
                              void* d_out, int out_size, void* d_ws, size_t ws_size,
                              hipStream_t stream) {
    const float*   h          = (const float*)d_in[0];   // (32,1024)
    const float*   inputs     = (const float*)d_in[1];   // (32,4000,512)
    const uint8_t* mask       = (const uint8_t*)d_in[2]; // (32,4000) bool
    const float*   mu         = (const float*)d_in[3];   // (32,1,5)
    const float*   Wq         = (const float*)d_in[4];   // (128,1024)
    const float*   bq         = (const float*)d_in[5];   // (128)
    const float*   Wv         = (const float*)d_in[6];   // (15,128)
    const float*   delta_bias = (const float*)d_in[7];   // (1,5)
    const float*   sigma_bias = (const float*)d_in[8];   // (1,5)
    float* out = (float*)d_out;                          // (32,512)

    char* ws = (char*)d_ws;
    float* params   = (float*)ws;                              // 2 KB
    float* att      = (float*)(ws + 4096);                     // 512 KB
    float* partials = (float*)(ws + 4096 + 512 * 1024);        // 2 MB

    gmm_params_kernel<<<1, 512, 0, stream>>>(h, Wq, bq, Wv, mu,
                                             delta_bias, sigma_bias, params);

    att_kernel<<<dim3((TLEN + 255) / 256, BSZ), 256, 0, stream>>>(mask, params, att);

    ctx_partial_kernel<<<dim3(TCHUNKS, BSZ), 128, 0, stream>>>(inputs, att, partials);

    ctx_reduce_kernel<<<BSZ, 512, 0, stream>>>(partials, out);
}